// EtaWeights_25151328485679
// MI455X (gfx1250) — compile-verified
//
#include <hip/hip_runtime.h>

// weight[i] = loss[i]*mask*eta   if loss[i] > eta
//           = -loss[i]/eta + 1   otherwise
//
// Memory-bound streaming kernel for MI455X:
//  - Exact-cover launch: grid*block*ITER == n4 (compile-time ITER), so the
//    hot path is 4 clustered global_load_b128 -> one wait -> VALU -> 4 NT
//    global_store_b128. No runtime trip-count division, no remainder loop.
//  - Input (128MB) fits in 192MB L2: RT loads keep it resident across
//    graph replays.
//  - Output is write-only: non-temporal stores (th:TH_NT) so the store
//    stream does not evict the L2-resident input.

typedef float f4 __attribute__((ext_vector_type(4)));

#define BLOCK 256
#define ITER  4

__device__ __forceinline__ f4 eta_map(f4 x, float e, float me, float inv_e) {
    f4 r;
    #pragma unroll
    for (int k = 0; k < 4; ++k) {
        float hi = x[k] * me;                             // loss > eta branch
        float lo = __builtin_fmaf(-x[k], inv_e, 1.0f);    // loss <= eta branch
        r[k] = (x[k] > e) ? hi : lo;                      // v_cndmask
    }
    return r;
}

// Exact-cover kernel: every thread does exactly ITER float4 elements,
// strided by the total thread count. No bounds checks -> loads cluster.
__global__ void __launch_bounds__(BLOCK)
eta_weights_exact(const float* __restrict__ loss,
                  const float* __restrict__ eta_p,
                  const float* __restrict__ mask_p,
                  float* __restrict__ out)
{
    const float e     = eta_p[0];
    const float m     = mask_p[0];
    const float me    = m * e;
    const float inv_e = 1.0f / e;

    const f4* __restrict__ in4  = (const f4*)loss;
    f4*       __restrict__ out4 = (f4*)out;

    const int tid    = blockIdx.x * BLOCK + threadIdx.x;
    const int stride = gridDim.x * BLOCK;

    // Issue all loads first (clustered b128 loads, max MLP)...
    f4 v[ITER];
    #pragma unroll
    for (int j = 0; j < ITER; ++j)
        v[j] = in4[tid + j * stride];

    // ...then compute + stream out with NT stores.
    #pragma unroll
    for (int j = 0; j < ITER; ++j) {
        f4 r = eta_map(v[j], e, me, inv_e);
        __builtin_nontemporal_store(r, &out4[tid + j * stride]);
    }
}

// Generic fallback (grid-stride, bounds-checked) for sizes that do not
// divide evenly. Never taken for the reference N = 2^25 but keeps
// kernel_launch correct for any input size.
__global__ void __launch_bounds__(BLOCK)
eta_weights_generic(const float* __restrict__ loss,
                    const float* __restrict__ eta_p,
                    const float* __restrict__ mask_p,
                    float* __restrict__ out,
                    int n)  // scalar element count
{
    const float e     = eta_p[0];
    const float m     = mask_p[0];
    const float me    = m * e;
    const float inv_e = 1.0f / e;

    const int n4     = n >> 2;
    const int stride = gridDim.x * BLOCK;
    const f4* __restrict__ in4  = (const f4*)loss;
    f4*       __restrict__ out4 = (f4*)out;

    for (int i = blockIdx.x * BLOCK + threadIdx.x; i < n4; i += stride) {
        f4 r = eta_map(in4[i], e, me, inv_e);
        __builtin_nontemporal_store(r, &out4[i]);
    }
    // scalar tail (n not a multiple of 4)
    int base = n4 << 2;
    for (int i = base + blockIdx.x * BLOCK + threadIdx.x; i < n; i += stride) {
        float x  = loss[i];
        float hi = x * me;
        float lo = __builtin_fmaf(-x, inv_e, 1.0f);
        out[i] = (x > e) ? hi : lo;
    }
}

extern "C" void kernel_launch(void* const* d_in, const int* in_sizes, int n_in,
                              void* d_out, int out_size, void* d_ws, size_t ws_size,
                              hipStream_t stream) {
    const float* loss = (const float*)d_in[0];
    const float* eta  = (const float*)d_in[1];
    const float* mask = (const float*)d_in[2];
    float* out        = (float*)d_out;

    const int n  = in_sizes[0];          // 2^25 for the reference
    const int n4 = n >> 2;               // float4 elements

    const int chunk = BLOCK * ITER;      // float4s covered per block
    if ((n & 3) == 0 && (n4 % chunk) == 0) {
        // Exact cover: n4 = grid * BLOCK * ITER.  For n=2^25: grid = 8192.
        const int grid = n4 / chunk;
        eta_weights_exact<<<grid, BLOCK, 0, stream>>>(loss, eta, mask, out);
    } else {
        int grid = (n4 + BLOCK - 1) / BLOCK;
        if (grid > 8192) grid = 8192;
        if (grid < 1) grid = 1;
        eta_weights_generic<<<grid, BLOCK, 0, stream>>>(loss, eta, mask, out, n);
    }
}